// FullMatch_80410377715994
// MI455X (gfx1250) — compile-verified
//
#include <hip/hip_runtime.h>
#include <math.h>

// Shapes fixed by the reference: v1[B,T,D], v1_mask[B,T], v2[B,D], w[L,D] -> out[B,T,L]
#define B_   16
#define T_   256
#define D_   512
#define L_   32
#define MT   32          // M-rows of output per block (2 waves x 16 rows)
#define EPS_ 1e-8f

typedef __attribute__((ext_vector_type(2))) float v2f;
typedef __attribute__((ext_vector_type(8))) float v8f;

// Block: 64 threads = 2 wave32 waves. Wave wm owns a 16(M) x 32(N) output slab:
// rows t0+wm*16..+15, all 32 L columns (two 16-wide WMMA tiles). Four f32 WMMA
// accumulators per wave:
//   cN0/cN1 = (mask*v1*v2) x w^2^T   (numerator, N-tiles 0 and 1)
//   cS0/cS1 = (mask*v1)^2  x w^2^T   (n1^2,      N-tiles 0 and 1)
// A-operand prep (3 pk_mul) is shared by 4 WMMAs -> 1.25 VALU ops per WMMA.
__global__ __launch_bounds__(64) void fullmatch_cosine_kernel(
    const float* __restrict__ v1,
    const float* __restrict__ v1_mask,
    const float* __restrict__ v2,
    const float* __restrict__ w,
    float* __restrict__ out)
{
    __shared__ float sV2[D_];    // v2[b,:]
    __shared__ float sMask[MT];  // mask rows for this tile
    __shared__ float sN2[L_];    // sum_d v2^2 * w^2 per l (pre-sqrt)

    const int tid = threadIdx.x;
    const int b   = blockIdx.x / (T_ / MT);
    const int t0  = (blockIdx.x % (T_ / MT)) * MT;

    // ---- stage v2[b,:] (float4, coalesced), mask rows, zero n2 accumulators ----
    {
        const float4* v2v = (const float4*)(v2 + (size_t)b * D_);
        ((float4*)sV2)[tid]      = v2v[tid];          // 64*4 = 256 floats
        ((float4*)sV2)[tid + 64] = v2v[tid + 64];     // next 256 floats
        if (tid < MT) sMask[tid] = v1_mask[(size_t)b * T_ + t0 + tid];
        if (tid < L_) sN2[tid]   = 0.0f;
    }
    __syncthreads();

    // ---- n2 partial sums: thread (l = tid&31, half q = tid>>5) covers 256 d's ----
    {
        const int l = tid & (L_ - 1);
        const int q = tid >> 5;                           // 0..1
        const float4* wr = (const float4*)(w + (size_t)l * D_ + q * 256);
        const float4* sv = (const float4*)(sV2 + q * 256);
        float acc = 0.0f;
        #pragma unroll 4
        for (int i = 0; i < 64; ++i) {
            float4 wv = wr[i];
            float4 vv = sv[i];
            float px = wv.x * vv.x, py = wv.y * vv.y;
            float pz = wv.z * vv.z, pw = wv.w * vv.w;
            acc += px * px + py * py + pz * pz + pw * pw; // (v2*w)^2 = v2^2 * w^2
        }
        atomicAdd(&sN2[l], acc);                          // ds_add_f32
    }

    // ---- per-wave tile setup ----
    const int lane = tid & 31;
    const int wm   = tid >> 5;                 // 0..1 : M sub-tile (16 rows each)
    const int hl   = lane & 15;
    const int koff = (lane >> 4) << 1;         // lanes 16-31 hold K = k+2,k+3

    const int   rowLocal = wm * 16 + hl;       // A row this lane feeds (M = hl)
    const float m        = sMask[rowLocal];
    const float* aRow    = v1 + ((size_t)b * T_ + t0 + rowLocal) * (size_t)D_;
    const float* wRow0   = w  + (size_t)(hl)      * (size_t)D_;   // B cols 0..15
    const float* wRow1   = w  + (size_t)(16 + hl) * (size_t)D_;   // B cols 16..31

    v8f cN0 = {}, cS0 = {};   // N-tile 0 accumulators (num, n1^2)
    v8f cN1 = {}, cS1 = {};   // N-tile 1 accumulators

    #pragma unroll 4
    for (int k = 0; k < D_; k += 4) {
        const int kb = k + koff;

        float2 av  = *(const float2*)(aRow  + kb);  // v1[b,row,kb..kb+1]
        float2 sv  = *(const float2*)(sV2   + kb);  // v2[b,kb..kb+1] (LDS broadcast)
        float2 wv0 = *(const float2*)(wRow0 + kb);  // w[col0,kb..kb+1]
        float2 wv1 = *(const float2*)(wRow1 + kb);  // w[col1,kb..kb+1]

        av.x *= m;  av.y *= m;                      // apply mask once

        v2f aN  = { av.x * sv.x, av.y * sv.y };     // (m*v1)*v2  — shared by 2 WMMAs
        v2f aS  = { av.x * av.x, av.y * av.y };     // (m*v1)^2   — shared by 2 WMMAs
        v2f bW0 = { wv0.x * wv0.x, wv0.y * wv0.y }; // w^2, N-tile 0
        v2f bW1 = { wv1.x * wv1.x, wv1.y * wv1.y }; // w^2, N-tile 1

        cN0 = __builtin_amdgcn_wmma_f32_16x16x4_f32(
                  false, aN, false, bW0, (short)0, cN0, false, false);
        cS0 = __builtin_amdgcn_wmma_f32_16x16x4_f32(
                  false, aS, false, bW0, (short)0, cS0, false, false);
        cN1 = __builtin_amdgcn_wmma_f32_16x16x4_f32(
                  false, aN, false, bW1, (short)0, cN1, false, false);
        cS1 = __builtin_amdgcn_wmma_f32_16x16x4_f32(
                  false, aS, false, bW1, (short)0, cS1, false, false);
    }

    __syncthreads();   // sN2 adds (pre-loop) are now visible

    // ---- epilogue: out = num / max(sqrt(n1^2) * n2, eps) ----
    const float n2v0  = sqrtf(sN2[hl]);                   // N-tile 0 column norm
    const float n2v1  = sqrtf(sN2[16 + hl]);              // N-tile 1 column norm
    const int   mBase = t0 + wm * 16 + ((lane >> 4) << 3); // lanes>=16 -> M = r+8

    #pragma unroll
    for (int r = 0; r < 8; ++r) {
        const size_t base = ((size_t)b * T_ + (size_t)(mBase + r)) * L_;
        float d0 = fmaxf(sqrtf(cS0[r]) * n2v0, EPS_);
        float d1 = fmaxf(sqrtf(cS1[r]) * n2v1, EPS_);
        out[base + hl]      = cN0[r] / d0;
        out[base + 16 + hl] = cN1[r] / d1;
    }
}

extern "C" void kernel_launch(void* const* d_in, const int* in_sizes, int n_in,
                              void* d_out, int out_size, void* d_ws, size_t ws_size,
                              hipStream_t stream) {
    const float* v1      = (const float*)d_in[0];
    const float* v1_mask = (const float*)d_in[1];
    const float* v2      = (const float*)d_in[2];
    const float* w       = (const float*)d_in[3];
    float*       out     = (float*)d_out;

    dim3 grid(B_ * (T_ / MT));   // 16 * 8 = 128 blocks
    dim3 block(64);              // 2 wave32 waves
    fullmatch_cosine_kernel<<<grid, block, 0, stream>>>(v1, v1_mask, v2, w, out);
}